// SpeakerLoss_13460427505706
// MI455X (gfx1250) — compile-verified
//
#include <hip/hip_runtime.h>
#include <hip/hip_bf16.h>
#include <math.h>

#define Bdim 16
#define Ndim 2
#define Ddim 512
#define Tdim 2000
#define Mdim 50
#define Mpad 64

typedef float v2f __attribute__((ext_vector_type(2)));
typedef float v8f __attribute__((ext_vector_type(8)));

// ---------------------------------------------------------------------------
// Pass 1: embedding repulsion regularizer + e2[m] + zero-padded E (64 x 512).
// Single block; tiny.
// ---------------------------------------------------------------------------
__global__ __launch_bounds__(1024) void reg_kernel(const float* __restrict__ E,
                                                   float* __restrict__ e2,
                                                   float* __restrict__ Epad,
                                                   float* __restrict__ reg_out) {
  __shared__ float mat[Mdim * Mdim];
  __shared__ float red[1024];
  const int tid = threadIdx.x;

  // zero-padded copy of E: rows [0,50) copied, rows [50,64) zeroed
  for (int p = tid; p < Mpad * Ddim; p += 1024) {
    const int row = p / Ddim;
    Epad[p] = (row < Mdim) ? E[p] : 0.f;
  }

  for (int p = tid; p < Mdim * Mdim; p += 1024) {
    const int i = p / Mdim, j = p % Mdim;
    const float* ei = E + (size_t)i * Ddim;
    const float* ej = E + (size_t)j * Ddim;
    float s = 0.f;
    for (int d = 0; d < Ddim; ++d) s += fabsf(ei[d] - ej[d]);
    mat[p] = s;
  }
  if (tid < Mdim) {
    const float* em = E + (size_t)tid * Ddim;
    float s = 0.f;
    for (int d = 0; d < Ddim; ++d) s += em[d] * em[d];
    e2[tid] = s;
  }
  __syncthreads();

  // global max of reg_mat
  float mx = -1e30f;
  for (int p = tid; p < Mdim * Mdim; p += 1024) mx = fmaxf(mx, mat[p]);
  red[tid] = mx;
  __syncthreads();
  for (int s = 512; s > 0; s >>= 1) {
    if (tid < s) red[tid] = fmaxf(red[tid], red[tid + s]);
    __syncthreads();
  }
  const float maxv = red[0];
  __syncthreads();

  // per-row min (diag + maxv), then -mean(log(min + eps))
  float lacc = 0.f;
  if (tid < Mdim) {
    float mn = 1e30f;
    for (int j = 0; j < Mdim; ++j) {
      float vv = mat[tid * Mdim + j] + ((j == tid) ? maxv : 0.f);
      mn = fminf(mn, vv);
    }
    lacc = __logf(mn + 1e-8f);
  }
  red[tid] = lacc;
  __syncthreads();
  for (int s = 512; s > 0; s >>= 1) {
    if (tid < s) red[tid] += red[tid + s];
    __syncthreads();
  }
  if (tid == 0) *reg_out = -red[0] * (1.f / Mdim);
}

// ---------------------------------------------------------------------------
// Pass 2: per-(b,n) fp32 WMMA GEMM  xe = Epad(64x512) * X(512xT), fused x2,
// softmin over m, and scatter of dplus at m=spkid[b,0], spkid[b,1].
// Block: 128 threads = 4 waves; wave w owns m-tile rows [16w, 16w+16).
// Grid: (125 t-tiles, 32 bn). Inner loop is branch-free (padded A rows).
// ---------------------------------------------------------------------------
__global__ __launch_bounds__(128) void gemm_softmin_kernel(
    const float* __restrict__ X,    // (B,N,D,T)
    const float* __restrict__ Epad, // (64,D) zero-padded
    const float* __restrict__ alpha,
    const float* __restrict__ beta,
    const int* __restrict__ spkid,  // (B,N)
    const float* __restrict__ e2,   // (50,) in ws
    float* __restrict__ D0,         // (B*N,T)
    float* __restrict__ D1) {       // (B*N,T)
  const int ttile = blockIdx.x;        // 0..124
  const int bn = blockIdx.y;           // 0..31
  const int b = bn >> 1;
  const int lane = threadIdx.x & 31;
  const int wave = threadIdx.x >> 5;   // m-tile index 0..3
  const int lhalf = lane >> 4;         // 0/1
  const int l16 = lane & 15;
  const int tg = ttile * 16 + l16;     // this lane's t column (B/C/D role)

  const float* Xb = X + (size_t)bn * Ddim * Tdim;

  // A-fragment row for this lane (A role): m = wave*16 + l16, K pair by lhalf
  const int mrowA = wave * 16 + l16;
  const float* __restrict__ Ea = Epad + (size_t)mrowA * Ddim + (lhalf * 2);
  const float* __restrict__ Xp = Xb + (size_t)(lhalf * 2) * Tdim + tg;

  v8f acc = {};
  float x2p = 0.f;
#pragma unroll 4
  for (int k = 0; k < Ddim; k += 4) {
    const v2f a = *(const v2f*)(Ea + k);            // b64, branch-free
    v2f bf;
    bf.x = Xp[(size_t)k * Tdim];
    bf.y = Xp[(size_t)(k + 1) * Tdim];
    x2p += bf.x * bf.x + bf.y * bf.y;
    // D = A(16x4 f32) * B(4x16 f32) + C
    acc = __builtin_amdgcn_wmma_f32_16x16x4_f32(false, a, false, bf,
                                                (short)0, acc, false, false);
  }

  // x2[t]: lane L covers d = {2*lhalf, 2*lhalf+1} mod 4; partner L^16 rest
  float x2t = x2p + __shfl_xor(x2p, 16, 32);

  const float scale = fabsf(alpha[0]) + 1e-5f;
  const float bet = beta[0];

  // C/D layout: m = wave*16 + lhalf*8 + v, t = l16
  float dist[8];
  float psum = 0.f;
#pragma unroll
  for (int v = 0; v < 8; ++v) {
    const int m = wave * 16 + lhalf * 8 + v;
    const float e2m = (m < Mdim) ? e2[m] : 0.f;
    const float fl = x2t + e2m - 2.f * acc[v];
    const float dd = scale * fl + bet;
    dist[v] = dd;
    psum += (m < Mdim) ? __expf(-dd) : 0.f;
  }
  // combine the two lane halves (same t, different m block)
  psum += __shfl_xor(psum, 16, 32);

  // cross-wave sum over the 4 m-tiles
  __shared__ float lds_ps[4][16];
  if (lhalf == 0) lds_ps[wave][l16] = psum;
  __syncthreads();
  const float tot = lds_ps[0][l16] + lds_ps[1][l16] + lds_ps[2][l16] + lds_ps[3][l16];
  const float others = __logf(tot * (1.f / Mdim) + 1e-8f);

  float dp[8];
#pragma unroll
  for (int v = 0; v < 8; ++v) dp[v] = dist[v] + others;

  // scatter dplus at the batch's two speaker ids
  const int s0 = spkid[b * 2 + 0];
  const int s1 = spkid[b * 2 + 1];
#pragma unroll 1
  for (int which = 0; which < 2; ++which) {
    const int s = which ? s1 : s0;
    if ((s >> 4) == wave && (((s >> 3) & 1) == lhalf)) {
      const int vsel = s & 7;
      float val = dp[0];
#pragma unroll
      for (int v = 1; v < 8; ++v)
        if (v == vsel) val = dp[v];
      float* Dst = which ? D1 : D0;
      Dst[(size_t)bn * Tdim + tg] = val;
    }
  }
}

// ---------------------------------------------------------------------------
// Pass 3: PIT over P=2 permutations; idx (first-on-tie like jnp.argmin),
// per-b loss partial sums.
// ---------------------------------------------------------------------------
__global__ __launch_bounds__(256) void pit_kernel(const float* __restrict__ D0,
                                                  const float* __restrict__ D1,
                                                  int* __restrict__ idxw,
                                                  float* __restrict__ loss_b) {
  const int b = blockIdx.x;
  const int tid = threadIdx.x;
  __shared__ float red[256];
  const float* D0a = D0 + (size_t)(b * 2 + 0) * Tdim;
  const float* D0b = D0 + (size_t)(b * 2 + 1) * Tdim;
  const float* D1a = D1 + (size_t)(b * 2 + 0) * Tdim;
  const float* D1b = D1 + (size_t)(b * 2 + 1) * Tdim;
  float acc = 0.f;
  for (int t = tid; t < Tdim; t += 256) {
    // perm0=(0,1): n0->s0, n1->s1 ; perm1=(1,0): n0->s1, n1->s0
    const float s0 = 0.5f * (D0a[t] + D1b[t]);
    const float s1 = 0.5f * (D1a[t] + D0b[t]);
    const int id = (s1 < s0) ? 1 : 0;
    idxw[b * Tdim + t] = id;
    acc += id ? s1 : s0;
  }
  red[tid] = acc;
  __syncthreads();
  for (int s = 128; s > 0; s >>= 1) {
    if (tid < s) red[tid] += red[tid + s];
    __syncthreads();
  }
  if (tid == 0) loss_b[b] = red[0];
}

// ---------------------------------------------------------------------------
// Pass 4: center = mean_t of permutation-gathered inputs; finalize loss.
// Grid (64, B), 256 threads = 8 waves; wave w handles d = blockIdx.x*8 + w.
// ---------------------------------------------------------------------------
__global__ __launch_bounds__(256) void center_kernel(const float* __restrict__ X,
                                                     const int* __restrict__ idxw,
                                                     const float* __restrict__ loss_b,
                                                     float* __restrict__ out) {
  const int b = blockIdx.y;
  const int wave = threadIdx.x >> 5;
  const int lane = threadIdx.x & 31;
  const int d = blockIdx.x * 8 + wave;
  const float* x0 = X + ((size_t)(b * 2 + 0) * Ddim + d) * Tdim;
  const float* x1 = X + ((size_t)(b * 2 + 1) * Ddim + d) * Tdim;
  const int* idp = idxw + (size_t)b * Tdim;
  float c0 = 0.f, c1 = 0.f;
  for (int t = lane; t < Tdim; t += 32) {
    const float a = x0[t], bb = x1[t];
    const int id = idp[t];
    c0 += id ? bb : a;   // h[b,0,d,t] = x[b, PERM[id,0], d, t]
    c1 += id ? a : bb;   // h[b,1,d,t] = x[b, PERM[id,1], d, t]
  }
#pragma unroll
  for (int s = 16; s > 0; s >>= 1) {
    c0 += __shfl_xor(c0, s, 32);
    c1 += __shfl_xor(c1, s, 32);
  }
  if (lane == 0) {
    out[1 + (size_t)(b * 2 + 0) * Ddim + d] = c0 * (1.f / Tdim);
    out[1 + (size_t)(b * 2 + 1) * Ddim + d] = c1 * (1.f / Tdim);
  }
  if (blockIdx.x == 0 && b == 0 && threadIdx.x == 0) {
    float s = 0.f;
    for (int i = 0; i < Bdim; ++i) s += loss_b[i];
    out[0] = s * (1.f / (Bdim * Tdim));
  }
}

// ---------------------------------------------------------------------------
extern "C" void kernel_launch(void* const* d_in, const int* in_sizes, int n_in,
                              void* d_out, int out_size, void* d_ws, size_t ws_size,
                              hipStream_t stream) {
  (void)in_sizes; (void)n_in; (void)out_size; (void)ws_size;
  const float* X     = (const float*)d_in[0];  // inputs (B,N,D,T)
  const float* alpha = (const float*)d_in[1];
  const float* beta  = (const float*)d_in[2];
  const float* E     = (const float*)d_in[3];  // emb_table (M,D)
  const int*   spk   = (const int*)d_in[4];    // spkid (B,N)
  float* out = (float*)d_out;                  // [loss(1), center(B*N*D), reg(1)]

  // workspace layout (floats)
  float* e2w    = (float*)d_ws;                // 64 (50 used)
  float* Epad   = e2w + 64;                    // 64*512
  float* D0w    = Epad + Mpad * Ddim;          // B*N*T
  float* D1w    = D0w + Bdim * Ndim * Tdim;    // B*N*T
  float* loss_b = D1w + Bdim * Ndim * Tdim;    // 16
  int*   idxw   = (int*)(loss_b + 16);         // B*T

  float* reg_out = out + 1 + Bdim * Ndim * Ddim;

  reg_kernel<<<1, 1024, 0, stream>>>(E, e2w, Epad, reg_out);
  gemm_softmin_kernel<<<dim3(Tdim / 16, Bdim * Ndim), 128, 0, stream>>>(
      X, Epad, alpha, beta, spk, e2w, D0w, D1w);
  pit_kernel<<<Bdim, 256, 0, stream>>>(D0w, D1w, idxw, loss_b);
  center_kernel<<<dim3(Ddim / 8, Bdim), 256, 0, stream>>>(X, idxw, loss_b, out);
}